// DCRNNet_33801392619882
// MI455X (gfx1250) — compile-verified
//
#include <hip/hip_runtime.h>
#include <math.h>

typedef float v2f __attribute__((ext_vector_type(2)));
typedef float v8f __attribute__((ext_vector_type(8)));

#define IN_CH   64
#define OUT_CH  128
#define NCOL    256   // Z preact (0..127) | Ht preact (128..255)
#define KORD    3
#define CAT     192   // gate input width in reference (only first 64 rows live)
#define OSZ     12

// ---------------- WMMA wrapper: fp32 16x16x4 ----------------
__device__ __forceinline__ v8f wmma4(v2f a, v2f b, v8f c) {
  return __builtin_amdgcn_wmma_f32_16x16x4_f32(false, a, false, b, (short)0, c, false, false);
}

// ---------------- utility kernels ----------------
__global__ void k_zero(float* p, long long count) {
  long long i = (long long)blockIdx.x * blockDim.x + threadIdx.x;
  if (i < count) p[i] = 0.0f;
}

__global__ void k_degree(const int* __restrict__ row, const int* __restrict__ col,
                         const float* __restrict__ w,
                         float* degO, float* degI, int nE) {
  int e = blockIdx.x * blockDim.x + threadIdx.x;
  if (e >= nE) return;
  float we = w[e];
  unsafeAtomicAdd(&degO[row[e]], we);
  unsafeAtomicAdd(&degI[col[e]], we);
}

__global__ void k_invert(float* p, int n) {
  int i = blockIdx.x * blockDim.x + threadIdx.x;
  if (i < n) p[i] = 1.0f / p[i];
}

// To[col] += invO[row] * srcO[row];  Ti[row] += invI[col] * srcI[col]
__global__ void k_prop(const float* __restrict__ srcO, const float* __restrict__ srcI,
                       float* To, float* Ti,
                       const int* __restrict__ row, const int* __restrict__ col,
                       const float* __restrict__ invO, const float* __restrict__ invI,
                       int nE) {
  int t = blockIdx.x * blockDim.x + threadIdx.x;
  int e = t >> 4;
  if (e >= nE) return;
  int c = (t & 15) << 2;
  int r = row[e], q = col[e];
  float so = invO[r];
  float si = invI[q];

  const float4 xo = *(const float4*)(srcO + (long long)r * IN_CH + c);
  float* po = To + (long long)q * IN_CH + c;
  unsafeAtomicAdd(po + 0, so * xo.x);
  unsafeAtomicAdd(po + 1, so * xo.y);
  unsafeAtomicAdd(po + 2, so * xo.z);
  unsafeAtomicAdd(po + 3, so * xo.w);

  const float4 xi = *(const float4*)(srcI + (long long)q * IN_CH + c);
  float* pi = Ti + (long long)r * IN_CH + c;
  unsafeAtomicAdd(pi + 0, si * xi.x);
  unsafeAtomicAdd(pi + 1, si * xi.y);
  unsafeAtomicAdd(pi + 2, si * xi.z);
  unsafeAtomicAdd(pi + 3, si * xi.w);
}

// T2 = 2*T2raw - X  (both directions)
__global__ void k_combine(float* T2o, float* T2i, const float* __restrict__ X, long long cnt) {
  long long i = (long long)blockIdx.x * blockDim.x + threadIdx.x;
  if (i < cnt) {
    float xv = X[i];
    T2o[i] = 2.0f * T2o[i] - xv;
    T2i[i] = 2.0f * T2i[i] - xv;
  }
}

// Pack stacked weights into WMMA B-fragment layout:
// logical Wcat[kk][j], kk in 0..319, j in 0..255; stored so each lane reads a
// contiguous v2f:  Wp[(k4*NCOL + j)*4 + (kk&3)] = Wcat[kk][j]
__global__ void k_pack(const float* __restrict__ Wz, const float* __restrict__ Wh,
                       float* __restrict__ Wp) {
  int idx = blockIdx.x * blockDim.x + threadIdx.x; // 320*256
  if (idx >= 320 * NCOL) return;
  int kk = idx / NCOL;
  int j  = idx % NCOL;
  const float* W = (j < OUT_CH) ? Wz : Wh;
  int jj = j & (OUT_CH - 1);
  int term = kk >> 6;   // 0:X  1:To  2:Ti  3:T2o  4:T2i
  int c = kk & 63;
  auto widx = [&](int d, int k) { return ((d * KORD + k) * CAT + c) * OUT_CH + jj; };
  float v;
  switch (term) {
    case 0:  v = W[widx(0, 0)] + W[widx(1, 0)]; break;
    case 1:  v = W[widx(0, 1)]; break;
    case 2:  v = W[widx(1, 1)]; break;
    case 3:  v = W[widx(0, 2)]; break;
    default: v = W[widx(1, 2)]; break;
  }
  int k4 = kk >> 2, rem = kk & 3;
  Wp[(k4 * NCOL + j) * 4 + rem] = v;
}

// ---------------- fused GEMM + gates + head ----------------
__device__ __forceinline__ void panel64(const float* __restrict__ S,
                                        const float* __restrict__ Wp,
                                        int kOff4, int mrow, int half2, int n0,
                                        v8f& a0, v8f& a1, v8f& a2, v8f& a3) {
  const float* sBase = S + (long long)mrow * IN_CH + half2;
#pragma unroll
  for (int kk = 0; kk < IN_CH; kk += 4) {
    v2f a = *(const v2f*)(sBase + kk);
    const float* wb = Wp + ((long long)(kOff4 + (kk >> 2)) * NCOL + n0) * 4 + half2;
    v2f b0 = *(const v2f*)(wb);
    a0 = wmma4(a, b0, a0);
    v2f b1 = *(const v2f*)(wb + 16 * 4);
    a1 = wmma4(a, b1, a1);
    v2f b2 = *(const v2f*)(wb + 32 * 4);
    a2 = wmma4(a, b2, a2);
    v2f b3 = *(const v2f*)(wb + 48 * 4);
    a3 = wmma4(a, b3, a3);
  }
}

__global__ __launch_bounds__(128)
void k_gemm(const float* __restrict__ X,  const float* __restrict__ To,
            const float* __restrict__ Ti, const float* __restrict__ T2o,
            const float* __restrict__ T2i, const float* __restrict__ Wp,
            const float* __restrict__ bz, const float* __restrict__ bh,
            const float* __restrict__ Wl, const float* __restrict__ bl,
            float* __restrict__ out, int n)
{
  __shared__ float P[16][NCOL + 8];
  const int m0   = blockIdx.x * 16;
  const int lane = threadIdx.x & 31;
  const int wave = threadIdx.x >> 5;
  const int half = lane >> 4;
  const int ln   = lane & 15;
  const int half2 = half * 2;
  const int n0   = wave * 64 + ln;

  int m = m0 + ln;
  if (m >= n) m = n - 1;

  v8f a0 = {0.f,0.f,0.f,0.f,0.f,0.f,0.f,0.f};
  v8f a1 = a0, a2 = a0, a3 = a0;

  panel64(X,   Wp, 0,  m, half2, n0, a0, a1, a2, a3);
  panel64(To,  Wp, 16, m, half2, n0, a0, a1, a2, a3);
  panel64(Ti,  Wp, 32, m, half2, n0, a0, a1, a2, a3);
  panel64(T2o, Wp, 48, m, half2, n0, a0, a1, a2, a3);
  panel64(T2i, Wp, 64, m, half2, n0, a0, a1, a2, a3);

  // C/D layout: VGPR i -> row (half?8+i:i), column n0 (+ tile*16)
#pragma unroll
  for (int i = 0; i < 8; i++) {
    int r = half * 8 + i;
    P[r][n0]      = a0[i];
    P[r][n0 + 16] = a1[i];
    P[r][n0 + 32] = a2[i];
    P[r][n0 + 48] = a3[i];
  }
  __syncthreads();

  // gates: H = (1 - sigmoid(z)) * tanh(h), then ReLU; thread j owns column j
  {
    int j = threadIdx.x; // 0..127
    float bzj = bz[j], bhj = bh[j];
#pragma unroll 4
    for (int mm = 0; mm < 16; mm++) {
      float z = P[mm][j] + bzj;
      float h = P[mm][j + OUT_CH] + bhj;
      float zs = 1.0f / (1.0f + __expf(-z));
      float g = (1.0f - zs) * tanhf(h);
      P[mm][j] = g > 0.0f ? g : 0.0f;
    }
  }
  __syncthreads();

  // head: (16 x 128) @ (128 x 12) + bl
  for (int idx = threadIdx.x; idx < 16 * OSZ; idx += 128) {
    int mm = idx / OSZ, o = idx - mm * OSZ;
    int gm = m0 + mm;
    if (gm >= n) continue;
    float s = bl[o];
#pragma unroll 8
    for (int c = 0; c < OUT_CH; c++) s += P[mm][c] * Wl[c * OSZ + o];
    out[(long long)gm * OSZ + o] = s;
  }
}

// ---------------- launcher ----------------
extern "C" void kernel_launch(void* const* d_in, const int* in_sizes, int n_in,
                              void* d_out, int out_size, void* d_ws, size_t ws_size,
                              hipStream_t stream) {
  const float* x  = (const float*)d_in[0];
  const int*   ei = (const int*)d_in[1];
  const float* ew = (const float*)d_in[2];
  const float* Wz = (const float*)d_in[3];
  const float* bz = (const float*)d_in[4];
  // d_in[5]=Wr, d_in[6]=br are dead (H0 == 0 -> R*H0 == 0)
  const float* Wh = (const float*)d_in[7];
  const float* bh = (const float*)d_in[8];
  const float* Wl = (const float*)d_in[9];
  const float* bl = (const float*)d_in[10];

  const int n  = in_sizes[0] / IN_CH;
  const int nE = in_sizes[2];
  const int* row = ei;
  const int* col = ei + nE;

  float* ws = (float*)d_ws;
  const long long nf = (long long)n * IN_CH;
  float* degO = ws;
  float* degI = degO + n;
  float* To   = degI + n;
  float* TiB  = To + nf;
  float* T2o  = TiB + nf;
  float* T2i  = T2o + nf;
  float* Wp   = T2i + nf;   // 320*256 floats

  const long long zcnt = 2LL * n + 4 * nf;
  k_zero<<<(unsigned)((zcnt + 255) / 256), 256, 0, stream>>>(ws, zcnt);
  k_pack<<<(320 * NCOL + 255) / 256, 256, 0, stream>>>(Wz, Wh, Wp);
  k_degree<<<(nE + 255) / 256, 256, 0, stream>>>(row, col, ew, degO, degI, nE);
  k_invert<<<(2 * n + 255) / 256, 256, 0, stream>>>(degO, 2 * n); // degO|degI contiguous

  const long long pt = (long long)nE * 16;
  k_prop<<<(unsigned)((pt + 255) / 256), 256, 0, stream>>>(x, x, To, TiB, row, col, degO, degI, nE);
  k_prop<<<(unsigned)((pt + 255) / 256), 256, 0, stream>>>(To, TiB, T2o, T2i, row, col, degO, degI, nE);
  k_combine<<<(unsigned)((nf + 255) / 256), 256, 0, stream>>>(T2o, T2i, x, nf);

  k_gemm<<<(n + 15) / 16, 128, 0, stream>>>(x, To, TiB, T2o, T2i, Wp,
                                            bz, bh, Wl, bl, (float*)d_out, n);
}